// PositionalEncoder_4844723110040
// MI455X (gfx1250) — compile-verified
//
#include <hip/hip_runtime.h>

// out[b,t,d] = x[b,t,d] + pe[t,d]
// x: (B=8, T=4096, D=1024) fp32. Pure streaming add: 256 MiB total traffic,
// bandwidth-bound (~11.5 us floor at 23.3 TB/s). One block per token row t;
// 256 threads (8 wave32) each own 4 consecutive dims, compute pe once, and
// reuse it across all 8 batches with b128 non-temporal loads/stores.

typedef float v4f __attribute__((ext_vector_type(4)));

#define PE_T 4096
#define PE_D 1024
#define PE_B 8

__global__ __launch_bounds__(256) void pe_add_kernel(const float* __restrict__ x,
                                                     float* __restrict__ out) {
    const int t   = blockIdx.x;        // token row 0..4095
    const int tid = threadIdx.x;       // 0..255
    const int d0  = tid << 2;          // first of 4 consecutive dims

    // pe for 4 consecutive dims; even i -> sin(angle), odd i -> cos(angle)
    // angle = t * 10000^(-i/D) = t * exp2(-(i/D) * log2(10000))
    const float pos = (float)t;
    const float NEG_L2_1E4 = -13.28771237954945f;  // -log2(10000)
    float pe[4];
#pragma unroll
    for (int k = 0; k < 4; ++k) {
        const int   i   = d0 + k;
        const float e   = (float)i * (1.0f / (float)PE_D);
        const float inv = exp2f(e * NEG_L2_1E4);   // 10000^(-i/D)
        const float ang = pos * inv;
        pe[k] = (i & 1) ? cosf(ang) : sinf(ang);
    }
    v4f pev;
    pev.x = pe[0]; pev.y = pe[1]; pev.z = pe[2]; pev.w = pe[3];

    const size_t rowBase     = (size_t)t * PE_D + (size_t)d0;
    const size_t batchStride = (size_t)PE_T * (size_t)PE_D;

    // Issue all 8 batch loads first (8 independent HBM streams in flight),
    // then add + store. Non-temporal: 256 MiB stream > 192 MB L2, no reuse.
    v4f v[PE_B];
#pragma unroll
    for (int b = 0; b < PE_B; ++b) {
        const v4f* src = (const v4f*)(x + rowBase + (size_t)b * batchStride);
        v[b] = __builtin_nontemporal_load(src);
    }
#pragma unroll
    for (int b = 0; b < PE_B; ++b) {
        v4f r = v[b] + pev;
        v4f* dst = (v4f*)(out + rowBase + (size_t)b * batchStride);
        __builtin_nontemporal_store(r, dst);
    }
}

extern "C" void kernel_launch(void* const* d_in, const int* in_sizes, int n_in,
                              void* d_out, int out_size, void* d_ws, size_t ws_size,
                              hipStream_t stream) {
    (void)in_sizes; (void)n_in; (void)d_ws; (void)ws_size; (void)out_size;
    const float* x   = (const float*)d_in[0];
    float*       out = (float*)d_out;

    dim3 grid(PE_T);    // one block per token row
    dim3 block(256);    // 8 wave32; D/4 float4 lanes
    hipLaunchKernelGGL(pe_add_kernel, grid, block, 0, stream, x, out);
}